// SpGraphAttentionLayer_29892972380479
// MI455X (gfx1250) — compile-verified
//
#include <hip/hip_runtime.h>

#define NODES 100000
#define EDGES 3200000
#define IN_F  256
#define OUT_F 128
#define LRELU_ALPHA 0.2f

typedef __attribute__((ext_vector_type(2))) float v2f;
typedef __attribute__((ext_vector_type(8))) float v8f;

// Monotone bijection float<->int so integer atomicMax == float max (non-NaN).
__device__ __forceinline__ int orderedFromFloat(float f) {
  int i = __float_as_int(f);
  return (i >= 0) ? i : (i ^ 0x7FFFFFFF);
}
__device__ __forceinline__ float floatFromOrdered(int i) {
  return __int_as_float((i >= 0) ? i : (i ^ 0x7FFFFFFF));
}

// ---------------------------------------------------------------------------
// 0) init: zero d_out (h_prime accumulator) and row_sum, set row_max = -inf
// ---------------------------------------------------------------------------
__global__ void gat_init(float* __restrict__ out, int* __restrict__ row_max,
                         float* __restrict__ row_sum) {
  size_t stride = (size_t)gridDim.x * blockDim.x;
  size_t i = (size_t)blockIdx.x * blockDim.x + threadIdx.x;
  const size_t total = (size_t)NODES * OUT_F;
  for (size_t j = i; j < total; j += stride) out[j] = 0.0f;
  for (size_t j = i; j < NODES; j += stride) {
    row_max[j] = orderedFromFloat(-3.0e38f);
    row_sum[j] = 0.0f;
  }
}

// ---------------------------------------------------------------------------
// 1) h = x @ W   via V_WMMA_F32_16X16X4_F32
//    Block = 256 threads = 8 waves; block owns a 16-row stripe of x (LDS),
//    wave w computes the 16x16 tile at columns [16w, 16w+16).
// ---------------------------------------------------------------------------
#define XS_STRIDE 260   // 256 + 4 pad: stride%64 != 0 -> conflict-free LDS reads

__global__ void __launch_bounds__(256, 2)
gat_gemm(const float* __restrict__ x, const float* __restrict__ W,
         float* __restrict__ h) {
  __shared__ __align__(16) float xs[16 * XS_STRIDE];
  const int tid = threadIdx.x;
  const int tile_row = blockIdx.x * 16;

  // Cooperative stage: 16 rows x 256 floats (16 KB) as float4, padded rows.
  {
    const float4* src = (const float4*)(x + (size_t)tile_row * IN_F);
    #pragma unroll
    for (int i = tid; i < 16 * (IN_F / 4); i += 256) {
      int row = i >> 6;       // /64 float4 per row
      int q   = i & 63;
      ((float4*)(xs + row * XS_STRIDE))[q] = src[(size_t)row * (IN_F / 4) + q];
    }
  }
  __syncthreads();

  const int wave = tid >> 5;
  const int lane = tid & 31;
  const int n    = lane & 15;   // N within tile; also M row for A frag
  const int hi   = lane >> 4;   // K-pair selector (0 -> K{0,1}, 1 -> K{2,3})
  const int col  = wave * 16 + n;

  // A frag (16x4 f32): lane<16 -> v0=K0,v1=K1 ; lane>=16 -> v0=K2,v1=K3
  const float* arow = xs + n * XS_STRIDE + hi * 2;
  // B frag (4x16 f32): mirrored K layout, N = lane&15
  const float* bptr = W + (size_t)(hi * 2) * OUT_F + col;

  v8f c = {0.f, 0.f, 0.f, 0.f, 0.f, 0.f, 0.f, 0.f};
  #pragma unroll 8
  for (int k0 = 0; k0 < IN_F; k0 += 4) {
    v2f a, b;
    a[0] = arow[k0];
    a[1] = arow[k0 + 1];
    b[0] = bptr[(size_t)k0 * OUT_F];
    b[1] = bptr[(size_t)(k0 + 1) * OUT_F];
    c = __builtin_amdgcn_wmma_f32_16x16x4_f32(false, a, false, b,
                                              (short)0, c, false, false);
  }

  // D layout: VGPR r -> (M = r + 8*hi, N = n)
  float* hout = h + (size_t)(tile_row + hi * 8) * OUT_F + col;
  #pragma unroll
  for (int r = 0; r < 8; ++r) hout[(size_t)r * OUT_F] = c[r];
}

// ---------------------------------------------------------------------------
// 2) s_src = h @ a[:128], s_dst = h @ a[128:]  (one wave per row)
// ---------------------------------------------------------------------------
__global__ void __launch_bounds__(256)
gat_scores(const float* __restrict__ h, const float* __restrict__ a,
           float* __restrict__ s_src, float* __restrict__ s_dst) {
  int row  = blockIdx.x * 8 + (threadIdx.x >> 5);
  int lane = threadIdx.x & 31;
  if (row >= NODES) return;
  float4 hv = ((const float4*)(h + (size_t)row * OUT_F))[lane];
  float4 as = ((const float4*)a)[lane];
  float4 ad = ((const float4*)(a + OUT_F))[lane];
  float ps = hv.x * as.x + hv.y * as.y + hv.z * as.z + hv.w * as.w;
  float pd = hv.x * ad.x + hv.y * ad.y + hv.z * ad.z + hv.w * ad.w;
  #pragma unroll
  for (int off = 16; off > 0; off >>= 1) {
    ps += __shfl_down(ps, off, 32);
    pd += __shfl_down(pd, off, 32);
  }
  if (lane == 0) { s_src[row] = ps; s_dst[row] = pd; }
}

// ---------------------------------------------------------------------------
// 3) per-edge leaky-relu score; segment-max by row (ordered-int atomicMax)
// ---------------------------------------------------------------------------
__global__ void __launch_bounds__(256)
gat_edge_max(const int* __restrict__ erow, const int* __restrict__ ecol,
             const float* __restrict__ s_src, const float* __restrict__ s_dst,
             int* __restrict__ row_max) {
  size_t i = (size_t)blockIdx.x * blockDim.x + threadIdx.x;
  if (i >= EDGES) return;
  int r = erow[i], cidx = ecol[i];
  float e = s_src[r] + s_dst[cidx];
  e = e > 0.0f ? e : LRELU_ALPHA * e;
  atomicMax(row_max + r, orderedFromFloat(e));
}

// ---------------------------------------------------------------------------
// 4) e_exp = exp(e - row_max[row]); segment-sum by row
// ---------------------------------------------------------------------------
__global__ void __launch_bounds__(256)
gat_edge_exp(const int* __restrict__ erow, const int* __restrict__ ecol,
             const float* __restrict__ s_src, const float* __restrict__ s_dst,
             const int* __restrict__ row_max, float* __restrict__ e_exp,
             float* __restrict__ row_sum) {
  size_t i = (size_t)blockIdx.x * blockDim.x + threadIdx.x;
  if (i >= EDGES) return;
  int r = erow[i], cidx = ecol[i];
  float e = s_src[r] + s_dst[cidx];
  e = e > 0.0f ? e : LRELU_ALPHA * e;
  float ex = __expf(e - floatFromOrdered(row_max[r]));
  e_exp[i] = ex;
  atomicAdd(row_sum + r, ex);
}

// ---------------------------------------------------------------------------
// 5) out[row] += (e_exp/row_sum[row]) * h[col]  — one wave per edge,
//    float4 gather per lane + 4 global f32 atomic adds (L2-resident).
// ---------------------------------------------------------------------------
__global__ void __launch_bounds__(256)
gat_aggregate(const int* __restrict__ erow, const int* __restrict__ ecol,
              const float* __restrict__ e_exp, const float* __restrict__ row_sum,
              const float* __restrict__ h, float* __restrict__ out) {
  size_t widx = (size_t)blockIdx.x * 8 + (threadIdx.x >> 5);
  if (widx >= EDGES) return;
  int lane = threadIdx.x & 31;
  int r = erow[widx], cidx = ecol[widx];
  float att = e_exp[widx] / row_sum[r];
  float4 v = ((const float4*)(h + (size_t)cidx * OUT_F))[lane];
  float* dst = out + (size_t)r * OUT_F + lane * 4;
  atomicAdd(dst + 0, att * v.x);
  atomicAdd(dst + 1, att * v.y);
  atomicAdd(dst + 2, att * v.z);
  atomicAdd(dst + 3, att * v.w);
}

// ---------------------------------------------------------------------------
// 6) in-place ELU on the accumulated output
// ---------------------------------------------------------------------------
__global__ void __launch_bounds__(256)
gat_elu(float* __restrict__ out) {
  size_t i = (size_t)blockIdx.x * blockDim.x + threadIdx.x;
  if (i < (size_t)NODES * OUT_F) {
    float v = out[i];
    out[i] = v > 0.0f ? v : (__expf(v) - 1.0f);
  }
}

// ---------------------------------------------------------------------------
extern "C" void kernel_launch(void* const* d_in, const int* in_sizes, int n_in,
                              void* d_out, int out_size, void* d_ws, size_t ws_size,
                              hipStream_t stream) {
  (void)in_sizes; (void)n_in; (void)out_size; (void)ws_size;
  const float* x    = (const float*)d_in[0];
  const int*   erow = (const int*)d_in[1];
  const int*   ecol = (const int*)d_in[2];
  const float* W    = (const float*)d_in[3];
  const float* a    = (const float*)d_in[4];
  float* out = (float*)d_out;

  // Workspace layout (floats): h | s_src | s_dst | row_max(int) | row_sum | e_exp
  float* ws      = (float*)d_ws;
  float* h       = ws;                              // NODES * OUT_F
  float* s_src   = h + (size_t)NODES * OUT_F;       // NODES
  float* s_dst   = s_src + NODES;                   // NODES
  int*   row_max = (int*)(s_dst + NODES);           // NODES
  float* row_sum = (float*)(row_max + NODES);       // NODES
  float* e_exp   = row_sum + NODES;                 // EDGES

  gat_init<<<2048, 256, 0, stream>>>(out, row_max, row_sum);
  gat_gemm<<<NODES / 16, 256, 0, stream>>>(x, W, h);                       // 6250 blocks
  gat_scores<<<(NODES + 7) / 8, 256, 0, stream>>>(h, a, s_src, s_dst);
  gat_edge_max<<<(EDGES + 255) / 256, 256, 0, stream>>>(erow, ecol, s_src, s_dst, row_max);
  gat_edge_exp<<<(EDGES + 255) / 256, 256, 0, stream>>>(erow, ecol, s_src, s_dst,
                                                        row_max, e_exp, row_sum);
  gat_aggregate<<<EDGES / 8, 256, 0, stream>>>(erow, ecol, e_exp, row_sum, h, out);
  gat_elu<<<((size_t)NODES * OUT_F + 255) / 256, 256, 0, stream>>>(out);
}